// PrimaryCapsuleLayer_82514911691249
// MI455X (gfx1250) — compile-verified
//
#include <hip/hip_runtime.h>
#include <math.h>

typedef __attribute__((ext_vector_type(2))) float v2f;
typedef __attribute__((ext_vector_type(8))) float v8f;

#define B_SZ   32
#define N_INP  16384
#define D_INP  8
#define JD     160              // N_OUT(10) * D_OUT(16)
#define NCHUNK 256
#define NPER   (N_INP / NCHUNK) // 64

__global__ void zero_kernel(float* p, int n) {
    int i = blockIdx.x * blockDim.x + threadIdx.x;
    if (i < n) p[i] = 0.0f;
}

// One workgroup = 20 waves; wave (mt,nt) owns C tile rows [mt*16,mt*16+16) x cols [nt*16,nt*16+16).
// Each workgroup reduces NPER capsules (n values) into per-chunk partial sums.
__global__ __launch_bounds__(640)
void route_pass(const float* __restrict__ x, const float* __restrict__ W,
                const float* __restrict__ Vcum,
                float* __restrict__ pE, float* __restrict__ pEU) {
    const int lane = threadIdx.x & 31;
    const int w    = threadIdx.x >> 5;   // 0..19
    const int mt   = w & 1;              // batch tile
    const int nt   = w >> 1;             // jd tile (0..9)
    const int chunk = blockIdx.x;

    const int hi = lane >> 4;            // 0/1
    const int lo = lane & 15;
    const int col     = nt * 16 + lo;            // jd column (0..159)
    const int rowBase = mt * 16 + hi * 8;        // C row base for this lane

    // Vcum values for the 8 C-layout rows this lane holds (constant across n-loop)
    float Vr[8];
#pragma unroll
    for (int r = 0; r < 8; ++r)
        Vr[r] = Vcum[(rowBase + r) * JD + col];

    // A operand: lane holds row M = mt*16+lo, K = hi*2 + {0,1} (VGPR0/1), step2 adds +4
    const float* ap = x + (size_t)(mt * 16 + lo) * (N_INP * D_INP) + hi * 2;
    // B operand: lane holds col = nt*16+lo, K = hi*2 + {0,1}, step2 adds +4 rows (stride JD)
    const float* wp = W + (size_t)(hi * 2) * JD + col;

    v8f accE = {};
    v8f accEU = {};
    const int n0 = chunk * NPER;
    for (int n = n0; n < n0 + NPER; ++n) {
        const float* a = ap + (size_t)n * D_INP;
        v2f a0 = *reinterpret_cast<const v2f*>(a);       // K = hi*2, hi*2+1
        v2f a1 = *reinterpret_cast<const v2f*>(a + 4);   // K = hi*2+4, hi*2+5
        const float* b = wp + (size_t)n * (D_INP * JD);
        v2f b0, b1;
        b0.x = b[0];
        b0.y = b[JD];
        b1.x = b[4 * JD];
        b1.y = b[5 * JD];
        v8f c = {};
        c = __builtin_amdgcn_wmma_f32_16x16x4_f32(false, a0, false, b0, (short)0, c, false, false);
        c = __builtin_amdgcn_wmma_f32_16x16x4_f32(false, a1, false, b1, (short)0, c, false, false);
#pragma unroll
        for (int r = 0; r < 8; ++r) {
            float u = c[r];                  // uhat[b = rowBase+r, jd = col] for this n
            float e = __expf(u * Vr[r]);     // iter0: Vcum==0 -> e==1 -> uniform softmax
            accE[r]  += e;
            accEU[r] += e * u;
        }
    }

    float* oE  = pE  + (size_t)chunk * (B_SZ * JD);
    float* oEU = pEU + (size_t)chunk * (B_SZ * JD);
#pragma unroll
    for (int r = 0; r < 8; ++r) {
        oE [(rowBase + r) * JD + col] = accE[r];
        oEU[(rowBase + r) * JD + col] = accEU[r];
    }
}

// One block per batch b; 160 threads, one per (j,d). Reduce chunk partials,
// s = sumEU/sumE, squash across d (groups of 16), update Vcum, optionally emit v.
__global__ __launch_bounds__(JD)
void finalize(const float* __restrict__ pE, const float* __restrict__ pEU,
              float* __restrict__ Vcum, float* __restrict__ out, int writeOut) {
    const int jd = threadIdx.x;   // 0..159
    const int b  = blockIdx.x;    // 0..31
    const size_t base = (size_t)b * JD + jd;

    float sE = 0.0f, sEU = 0.0f;
    for (int c = 0; c < NCHUNK; ++c) {
        sE  += pE [(size_t)c * (B_SZ * JD) + base];
        sEU += pEU[(size_t)c * (B_SZ * JD) + base];
    }
    float s = sEU / sE;

    __shared__ float sh[JD];
    sh[jd] = s;
    __syncthreads();

    const int j0 = (jd >> 4) << 4;
    float norm = 0.0f;
#pragma unroll
    for (int d = 0; d < 16; ++d) {
        float t = sh[j0 + d];
        norm += t * t;
    }
    // squash: v = |s|^2 * s / ((1 + |s|^2) * |s|)
    float v = norm * s / ((1.0f + norm) * sqrtf(norm));

    Vcum[base] += v;
    if (writeOut) out[base] = v;
}

extern "C" void kernel_launch(void* const* d_in, const int* in_sizes, int n_in,
                              void* d_out, int out_size, void* d_ws, size_t ws_size,
                              hipStream_t stream) {
    const float* x = (const float*)d_in[0];   // [32, 16384, 8]
    const float* W = (const float*)d_in[1];   // [16384, 8, 10, 16]
    float* out = (float*)d_out;               // [32, 10, 16]
    float* ws  = (float*)d_ws;

    float* Vcum = ws;                                   // 5120 floats
    float* pE   = Vcum + B_SZ * JD;                     // NCHUNK * 5120
    float* pEU  = pE + (size_t)NCHUNK * B_SZ * JD;      // NCHUNK * 5120

    zero_kernel<<<(B_SZ * JD + 255) / 256, 256, 0, stream>>>(Vcum, B_SZ * JD);

    for (int it = 0; it < 3; ++it) {
        route_pass<<<NCHUNK, 640, 0, stream>>>(x, W, Vcum, pE, pEU);
        finalize<<<B_SZ, JD, 0, stream>>>(pE, pEU, Vcum, out, it == 2 ? 1 : 0);
    }
}